// FieldAwareFactorizationMachine_53437983097346
// MI455X (gfx1250) — compile-verified
//
#include <hip/hip_runtime.h>
#include <hip/hip_bf16.h>
#include <math.h>

// FFM: out[b] = sigmoid( sum_{i<j} <T_i[adj_j], T_j[adj_i]> + sum_f w[adj_f] + bias )
//
// 1 wave (wave32) per sample. The 190 i<j pair dot-products are the trace of a
// 16x16 (K=192) fp32 matmul computed with V_WMMA_F32_16X16X4_F32:
//   A chunk (16x4): lane d holds element d of W[i_p][j_p]  (K = pair index)
//   B chunk (4x16): lane d holds element d of W[j_p][i_p]
//   D[d,d] accumulates sum_p W[i][j][d] * W[j][i][d] -> trace = FFM interaction.
//
// Memory-bound gather (~200 MB random 64B rows from a 1.28 GB table, ~9 us floor
// @ 23.3 TB/s). All 384 gather byte-offsets (fit in u32) are precomputed per wave
// into LDS. The hot loop is group double-buffered (WMMAs of group g overlap the
// 16 in-flight gathers of group g+1 -> partial s_wait_loadcnt), and each group
// computes all 16 addresses into distinct registers BEFORE issuing the 16 loads,
// so no address VGPR is overwritten while its load is in flight (kills the
// per-load s_wait_xcnt stalls).

typedef __attribute__((ext_vector_type(2))) float v2f;
typedef __attribute__((ext_vector_type(8))) float v8f;

#define NF 20
#define ED 16
#define NPAIR 190        // 20 choose 2
#define NPAIR_PAD 192    // pad to multiple of 4 (WMMA K)
#define NCHUNK 48        // 192 / 4
#define NSLOT 384        // 2 offsets (A,B) per padded pair
#define WAVES_PER_BLOCK 8
#define PF 4             // chunks per double-buffered group
#define NG (NCHUNK / PF) // 12 groups
static constexpr unsigned TVu = 1000001u;   // total vocab (20*50000 + 1)

__device__ __forceinline__ float ldg_off(const float* __restrict__ base, unsigned byteOff) {
    // uniform SGPR base + zext(u32) per-lane offset -> global_load_b32 saddr form
    return *(const float*)((const char*)base + (unsigned long long)byteOff);
}

__device__ __forceinline__ void load_group(const float* __restrict__ tables,
                                           const unsigned* soff, int cbase, int h, unsigned d4,
                                           float (&a0)[PF], float (&a1)[PF],
                                           float (&b0)[PF], float (&b1)[PF]) {
    uint4 o[PF];
    #pragma unroll
    for (int k = 0; k < PF; ++k)
        o[k] = *(const uint4*)&soff[8 * (cbase + k) + 4 * h];   // ds_load_b128 x4
    // Phase 1: all 16 voffsets into distinct live registers.
    unsigned va[PF * 4];
    #pragma unroll
    for (int k = 0; k < PF; ++k) {
        va[4 * k + 0] = o[k].x + d4;
        va[4 * k + 1] = o[k].y + d4;
        va[4 * k + 2] = o[k].z + d4;
        va[4 * k + 3] = o[k].w + d4;
    }
    // Phase 2: 16 back-to-back gathers (no address reg is recycled mid-flight).
    #pragma unroll
    for (int k = 0; k < PF; ++k) {
        a0[k] = ldg_off(tables, va[4 * k + 0]);
        b0[k] = ldg_off(tables, va[4 * k + 1]);
        a1[k] = ldg_off(tables, va[4 * k + 2]);
        b1[k] = ldg_off(tables, va[4 * k + 3]);
    }
}

__global__ __launch_bounds__(256)
void FieldAwareFactorizationMachine_53437983097346_kernel(
        const int* __restrict__ x,
        const float* __restrict__ tables,
        const float* __restrict__ linw,
        const float* __restrict__ linb,
        float* __restrict__ out, int batch)
{
    // Block-shared constant pair table: (i<<8)|j for the 190 i<j pairs.
    __shared__ unsigned short s_pair[NPAIR_PAD];
    // Per-wave gather byte-offset table. Slot 2p = A-side (row i*TV+adj[j]),
    // slot 2p+1 = B-side (row j*TV+adj[i]); offsets are row*64 bytes. Pad -> 0
    // (a valid in-bounds address; its contribution is masked to zero later).
    __shared__ __align__(16) unsigned s_off[WAVES_PER_BLOCK][NSLOT];

    const int t = threadIdx.x;
    if (t < NPAIR_PAD) {
        unsigned short v = 0;
        if (t < NPAIR) {
            int idx = t, i = 0, cnt = NF - 1;
            while (idx >= cnt) { idx -= cnt; ++i; --cnt; }
            int j = i + 1 + idx;
            v = (unsigned short)((i << 8) | j);
        }
        s_pair[t] = v;
    }
    __syncthreads();

    const int lane = t & 31;
    const int wave = t >> 5;
    const int b = blockIdx.x * WAVES_PER_BLOCK + wave;
    if (b >= batch) return;   // wave-uniform

    // Lane f (< 20) holds adjusted index for field f: x + f*50000.
    int adjv = 0;
    if (lane < NF) adjv = x[b * NF + lane] + lane * 50000;

    // ---- Setup: build all 384 gather offsets for this wave (12 slots/lane) ----
    unsigned* soff = &s_off[wave][0];
    #pragma unroll
    for (int q = 0; q < NSLOT / 32; ++q) {
        const int s = q * 32 + lane;              // slot 0..383
        const int p = s >> 1;                     // padded pair index
        const int pc = (p < NPAIR) ? p : 0;       // clamp (keeps EXEC full for shfl)
        const unsigned pr = s_pair[pc];
        const int i = (pr >> 8) & 0xFF, j = pr & 0xFF;
        const int aj = __shfl(adjv, j, 32);
        const int ai = __shfl(adjv, i, 32);
        const unsigned row = (s & 1) ? ((unsigned)j * TVu + (unsigned)ai)   // B: W[j][i]
                                     : ((unsigned)i * TVu + (unsigned)aj);  // A: W[i][j]
        soff[s] = (p < NPAIR) ? (row << 6) : 0u;  // row * 64 bytes
    }

    const int h = lane >> 4;                            // half-wave: K0/K1 vs K2/K3
    const unsigned d4 = (unsigned)((lane & 15) << 2);   // element byte offset
    const float hmask = h ? 0.f : 1.f;                  // zeroes pad pairs 190,191

    // ---- Hot loop: group double-buffered gathers feeding WMMA ----
    float A0[2][PF], A1[2][PF], B0[2][PF], B1[2][PF];
    load_group(tables, soff, 0, h, d4, A0[0], A1[0], B0[0], B1[0]);

    v8f acc = {};
    #pragma unroll
    for (int g = 0; g < NG; ++g) {
        const int cur = g & 1, nxt = cur ^ 1;
        if (g + 1 < NG)
            load_group(tables, soff, (g + 1) * PF, h, d4,
                       A0[nxt], A1[nxt], B0[nxt], B1[nxt]);
        #pragma unroll
        for (int k = 0; k < PF; ++k) {
            float a0 = A0[cur][k], a1 = A1[cur][k];
            float b0 = B0[cur][k], b1 = B1[cur][k];
            if (g == NG - 1 && k == PF - 1) {     // last chunk holds pad pairs on h==1
                a0 *= hmask; a1 *= hmask; b0 *= hmask; b1 *= hmask;
            }
            v2f av; av.x = a0; av.y = a1;
            v2f bv; bv.x = b0; bv.y = b1;
            acc = __builtin_amdgcn_wmma_f32_16x16x4_f32(
                    false, av, false, bv, (short)0, acc, false, false);
        }
    }

    // Trace of D: diag element M=r lives at (VGPR r, lane r); M=r+8 at (VGPR r, lane r+24).
    float val = 0.f;
    #pragma unroll
    for (int r = 0; r < 8; ++r) {
        if (lane == r)      val += acc[r];
        if (lane == r + 24) val += acc[r];
    }
    // Linear term: lane f adds w[adj_f].
    if (lane < NF) val += linw[adjv];

    // Wave32 butterfly reduction.
    #pragma unroll
    for (int off = 16; off; off >>= 1) val += __shfl_xor(val, off, 32);

    if (lane == 0) {
        float z = val + linb[0];
        out[b] = 1.0f / (1.0f + expf(-z));
    }
}

extern "C" void kernel_launch(void* const* d_in, const int* in_sizes, int n_in,
                              void* d_out, int out_size, void* d_ws, size_t ws_size,
                              hipStream_t stream) {
    const int*   x      = (const int*)  d_in[0];  // (BATCH, 20) int32
    const float* tables = (const float*)d_in[1];  // (20, 1000001, 16) f32
    const float* linw   = (const float*)d_in[2];  // (1000001, 1) f32
    const float* linb   = (const float*)d_in[3];  // (1,) f32
    float* out = (float*)d_out;                   // (BATCH,) f32

    const int batch = in_sizes[0] / NF;
    dim3 grid((batch + WAVES_PER_BLOCK - 1) / WAVES_PER_BLOCK), block(256);
    hipLaunchKernelGGL(FieldAwareFactorizationMachine_53437983097346_kernel,
                       grid, block, 0, stream,
                       x, tables, linw, linb, out, batch);
}